// DependencyParseModel_75754633166937
// MI455X (gfx1250) — compile-verified
//
#include <hip/hip_runtime.h>
#include <hip/hip_bf16.h>

#define SEQ   320
#define HID   400
#define INDIM 400
#define BI    800
#define GATES 1600
#define MLPN  1600
#define OUTN  321

typedef float v2f __attribute__((ext_vector_type(2)));
typedef float v8f __attribute__((ext_vector_type(8)));

// -------------------------------------------------------------------------
// Embedding gather: X[t] = concat(word_emb[words[t]](300), tag_emb[tags[t]](100))
// -------------------------------------------------------------------------
__global__ void embed_kernel(const int* __restrict__ words,
                             const int* __restrict__ tags,
                             const float* __restrict__ wemb,
                             const float* __restrict__ temb,
                             float* __restrict__ X) {
    int t = blockIdx.x;
    int w = words[t];
    int g = tags[t];
    for (int k = threadIdx.x; k < INDIM; k += blockDim.x) {
        float v = (k < 300) ? wemb[w * 300 + k] : temb[g * 100 + (k - 300)];
        X[t * INDIM + k] = v;
    }
}

// -------------------------------------------------------------------------
// WMMA f32 GEMM:  C[m, coff+n] = sum_k A[m,k] * W[n, woff+k] (+bias0[n]+bias1[n])
// One wave (32 threads) per 16x16 C tile, K stepped by 4 via v_wmma_f32_16x16x4_f32.
// Per ISA 7.12.2:
//   A 16x4 : lane L (m=L&15): VGPR0=A[m][kb+kh], VGPR1=A[m][kb+kh+1], kh = (L>=16)?2:0
//   B 4x16 : lane L (n=L&15): VGPR0=B[kh][n]=W[n][kb+kh], VGPR1=W[n][kb+kh+1]
//   C 16x16: VGPR r: lanes 0-15 -> row r, lanes 16-31 -> row r+8, col = L&15
// -------------------------------------------------------------------------
__global__ void gemm_xwT_wmma(const float* __restrict__ A, int lda,
                              const float* __restrict__ W, int ldw, int woff,
                              const float* __restrict__ bias0,
                              const float* __restrict__ bias1,
                              float* __restrict__ C, int ldc, int coff,
                              int K) {
    const int lane = threadIdx.x;           // 0..31, one full wave
    const int m0 = blockIdx.y * 16;
    const int n0 = blockIdx.x * 16;
    const int r15 = lane & 15;
    const int kh = (lane >> 4) ? 2 : 0;     // K half offset within the 4-slice

    const float* arow = A + (size_t)(m0 + r15) * lda + kh;
    const float* wrow = W + (size_t)(n0 + r15) * ldw + woff + kh;

    v8f acc = {};
    for (int kb = 0; kb < K; kb += 4) {
        v2f av, bv;
        av.x = arow[kb];  av.y = arow[kb + 1];
        bv.x = wrow[kb];  bv.y = wrow[kb + 1];
        acc = __builtin_amdgcn_wmma_f32_16x16x4_f32(
            /*neg_a=*/false, av, /*neg_b=*/false, bv,
            /*c_mod=*/(short)0, acc, /*reuse_a=*/false, /*reuse_b=*/false);
    }

    float bsum = 0.0f;
    if (bias0) bsum += bias0[n0 + r15];
    if (bias1) bsum += bias1[n0 + r15];

    const int mbase = (lane >> 4) ? 8 : 0;
    float* crow = C + (size_t)(m0 + mbase) * ldc + coff + n0 + r15;
    #pragma unroll
    for (int r = 0; r < 8; ++r) {
        crow[(size_t)r * ldc] = acc[r] + bsum;
    }
}

// -------------------------------------------------------------------------
// Sequential BiLSTM layer: grid.x = 2 (0 = forward, 1 = backward).
// Gin holds precomputed x@W_ih.T + b_ih + b_hh, shape (SEQ, 1600) per direction.
// Recurrent GEMV h@W_hh.T done in-workgroup with h/c/g in LDS.
// -------------------------------------------------------------------------
__device__ __forceinline__ float sigmoidf(float x) {
    return 1.0f / (1.0f + expf(-x));
}

__global__ void lstm_layer_kernel(const float* __restrict__ Gin_f,
                                  const float* __restrict__ Gin_b,
                                  const float* __restrict__ Whh_f,
                                  const float* __restrict__ Whh_b,
                                  const float* __restrict__ h0,  // (4, HID)
                                  const float* __restrict__ c0,  // (4, HID)
                                  int state_row_f,               // row in h0/c0 for fwd
                                  float* __restrict__ out,       // (SEQ, 2*HID)
                                  int out_stride) {
    const int dir = blockIdx.x;              // 0 fwd, 1 bwd
    const float* Gin = dir ? Gin_b : Gin_f;
    const float* Whh = dir ? Whh_b : Whh_f;
    const float* h_init = h0 + (size_t)(state_row_f + dir) * HID;
    const float* c_init = c0 + (size_t)(state_row_f + dir) * HID;
    const int col_off = dir * HID;

    __shared__ __align__(16) float h_s[HID];
    __shared__ __align__(16) float c_s[HID];
    __shared__ __align__(16) float g_s[GATES];

    for (int u = threadIdx.x; u < HID; u += blockDim.x) {
        h_s[u] = h_init[u];
        c_s[u] = c_init[u];
    }

    for (int s = 0; s < SEQ; ++s) {
        const int t = dir ? (SEQ - 1 - s) : s;
        __syncthreads();   // h_s/c_s from previous step visible

        // gates: g[j] = Gin[t][j] + sum_k h_s[k] * Whh[j][k]
        const float* gin_row = Gin + (size_t)t * GATES;
        for (int j = threadIdx.x; j < GATES; j += blockDim.x) {
            const float4* wr = reinterpret_cast<const float4*>(Whh + (size_t)j * HID);
            const float4* hv = reinterpret_cast<const float4*>(h_s);
            __builtin_prefetch(Whh + (size_t)(j + blockDim.x) * HID, 0, 1);
            float acc = gin_row[j];
            #pragma unroll 4
            for (int k4 = 0; k4 < HID / 4; ++k4) {
                float4 wv = wr[k4];
                float4 hvv = hv[k4];
                acc += hvv.x * wv.x + hvv.y * wv.y + hvv.z * wv.z + hvv.w * wv.w;
            }
            g_s[j] = acc;
        }
        __syncthreads();

        // cell/hidden update: split order i, f, gc, o
        for (int u = threadIdx.x; u < HID; u += blockDim.x) {
            float ig = sigmoidf(g_s[u]);
            float fg = sigmoidf(g_s[HID + u]);
            float gc = tanhf(g_s[2 * HID + u]);
            float og = sigmoidf(g_s[3 * HID + u]);
            float cn = fg * c_s[u] + ig * gc;
            c_s[u] = cn;
            float hn = og * tanhf(cn);
            h_s[u] = hn;
            out[(size_t)t * out_stride + col_off + u] = hn;
        }
    }
}

// -------------------------------------------------------------------------
// Pairwise scoring: out[1+i][1+j] = (i==j) ? 0 :
//     sum_k relu(Aa[i,k] + Bb[j,k]) * w2[k] + b2     (mlp_b1 folded into Aa)
// 16x16 pair tiles per block (256 threads), K chunked through LDS.
// Tiles are staged memory->LDS with CDNA5 GLOBAL_LOAD_ASYNC_TO_LDS_B128
// (ASYNCcnt-tracked, no VGPR round-trip), synchronized with s_wait_asynccnt.
// -------------------------------------------------------------------------
#define KCHUNK 160   // 1600 / 160 = 10 chunks; 160/4 = 40 float4 per row

__device__ __forceinline__ void async_copy_b128(const float* gsrc, float* lds_dst) {
    unsigned lds_off = (unsigned)(size_t)lds_dst;   // addrspace(3) byte offset
    asm volatile("global_load_async_to_lds_b128 %0, %1, off"
                 :: "v"(lds_off), "v"(gsrc) : "memory");
}

__global__ void score_kernel(const float* __restrict__ Aa,
                             const float* __restrict__ Bb,
                             const float* __restrict__ w2,
                             const float* __restrict__ b2,
                             float* __restrict__ out) {
    __shared__ __align__(16) float aT[16][KCHUNK];
    __shared__ __align__(16) float bT[16][KCHUNK];
    __shared__ __align__(16) float wL[KCHUNK];

    const int i0 = blockIdx.y * 16;
    const int j0 = blockIdx.x * 16;
    const int ti = threadIdx.x >> 4;
    const int tj = threadIdx.x & 15;
    const int KC4 = KCHUNK / 4;              // float4 per row

    float acc = 0.0f;
    for (int kb = 0; kb < MLPN; kb += KCHUNK) {
        __syncthreads();                     // previous chunk fully consumed
        // async stage: 16 rows x 40 float4 per matrix, 256 threads
        for (int idx = threadIdx.x; idx < 16 * KC4; idx += 256) {
            int r = idx / KC4, c4 = idx % KC4;
            async_copy_b128(&Aa[(size_t)(i0 + r) * MLPN + kb + c4 * 4], &aT[r][c4 * 4]);
            async_copy_b128(&Bb[(size_t)(j0 + r) * MLPN + kb + c4 * 4], &bT[r][c4 * 4]);
        }
        if (threadIdx.x < KC4) {
            async_copy_b128(&w2[kb + threadIdx.x * 4], &wL[threadIdx.x * 4]);
        }
        asm volatile("s_wait_asynccnt 0x0" ::: "memory");
        __syncthreads();                     // LDS tiles visible to all waves

        #pragma unroll 4
        for (int k = 0; k < KCHUNK; ++k) {
            acc += fmaxf(aT[ti][k] + bT[tj][k], 0.0f) * wL[k];
        }
    }

    const int i = i0 + ti, j = j0 + tj;
    float v = (i == j) ? 0.0f : (acc + b2[0]);
    out[(size_t)(1 + i) * OUTN + (1 + j)] = v;
}

// Row 0 / col 0 border, out[0][0] = 1
__global__ void border_kernel(float* __restrict__ out) {
    int t = blockIdx.x * blockDim.x + threadIdx.x;
    if (t < OUTN) out[t] = (t == 0) ? 1.0f : 0.0f;          // row 0
    if (t >= 1 && t < OUTN) out[(size_t)t * OUTN] = 0.0f;   // col 0
}

// -------------------------------------------------------------------------
// Host-side orchestration
// -------------------------------------------------------------------------
extern "C" void kernel_launch(void* const* d_in, const int* in_sizes, int n_in,
                              void* d_out, int out_size, void* d_ws, size_t ws_size,
                              hipStream_t stream) {
    const int*   words    = (const int*)d_in[0];
    const int*   tags     = (const int*)d_in[1];
    // d_in[2] = arcs (unused by reference)
    const float* word_emb = (const float*)d_in[3];
    const float* tag_emb  = (const float*)d_in[4];
    const float* h0       = (const float*)d_in[5];
    const float* c0       = (const float*)d_in[6];
    const float* w_ih_l0  = (const float*)d_in[7];
    const float* w_hh_l0  = (const float*)d_in[8];
    const float* b_ih_l0  = (const float*)d_in[9];
    const float* b_hh_l0  = (const float*)d_in[10];
    const float* w_ih_l0r = (const float*)d_in[11];
    const float* w_hh_l0r = (const float*)d_in[12];
    const float* b_ih_l0r = (const float*)d_in[13];
    const float* b_hh_l0r = (const float*)d_in[14];
    const float* w_ih_l1  = (const float*)d_in[15];
    const float* w_hh_l1  = (const float*)d_in[16];
    const float* b_ih_l1  = (const float*)d_in[17];
    const float* b_hh_l1  = (const float*)d_in[18];
    const float* w_ih_l1r = (const float*)d_in[19];
    const float* w_hh_l1r = (const float*)d_in[20];
    const float* b_ih_l1r = (const float*)d_in[21];
    const float* b_hh_l1r = (const float*)d_in[22];
    const float* mlp_w1   = (const float*)d_in[23];
    const float* mlp_b1   = (const float*)d_in[24];
    const float* mlp_w2   = (const float*)d_in[25];
    const float* mlp_b2   = (const float*)d_in[26];

    float* ws   = (float*)d_ws;
    float* X    = ws;                         // 320*400
    float* Gf   = X    + SEQ * INDIM;         // 320*1600
    float* Gb   = Gf   + SEQ * GATES;         // 320*1600
    float* out0 = Gb   + SEQ * GATES;         // 320*800
    float* hv   = out0 + SEQ * BI;            // 320*800
    float* Aa   = hv   + SEQ * BI;            // 320*1600
    float* Bb   = Aa   + SEQ * MLPN;          // 320*1600
    float* outf = (float*)d_out;

    // 1) embeddings
    embed_kernel<<<SEQ, 128, 0, stream>>>(words, tags, word_emb, tag_emb, X);

    // 2) layer 0 input-gate GEMMs (WMMA): (320x400)@(400x1600)
    {
        dim3 grid(GATES / 16, SEQ / 16);
        gemm_xwT_wmma<<<grid, 32, 0, stream>>>(X, INDIM, w_ih_l0,  INDIM, 0,
                                               b_ih_l0,  b_hh_l0,  Gf, GATES, 0, INDIM);
        gemm_xwT_wmma<<<grid, 32, 0, stream>>>(X, INDIM, w_ih_l0r, INDIM, 0,
                                               b_ih_l0r, b_hh_l0r, Gb, GATES, 0, INDIM);
    }

    // 3) layer 0 recurrence (fwd + bwd)
    lstm_layer_kernel<<<2, 512, 0, stream>>>(Gf, Gb, w_hh_l0, w_hh_l0r,
                                             h0, c0, /*state_row_f=*/0,
                                             out0, BI);

    // 4) layer 1 input-gate GEMMs: (320x800)@(800x1600)
    {
        dim3 grid(GATES / 16, SEQ / 16);
        gemm_xwT_wmma<<<grid, 32, 0, stream>>>(out0, BI, w_ih_l1,  BI, 0,
                                               b_ih_l1,  b_hh_l1,  Gf, GATES, 0, BI);
        gemm_xwT_wmma<<<grid, 32, 0, stream>>>(out0, BI, w_ih_l1r, BI, 0,
                                               b_ih_l1r, b_hh_l1r, Gb, GATES, 0, BI);
    }

    // 5) layer 1 recurrence
    lstm_layer_kernel<<<2, 512, 0, stream>>>(Gf, Gb, w_hh_l1, w_hh_l1r,
                                             h0, c0, /*state_row_f=*/2,
                                             hv, BI);

    // 6) MLP projections (WMMA): a = hv@w1a.T + mlp_b1 ; b = hv@w1b.T
    {
        dim3 grid(MLPN / 16, SEQ / 16);
        gemm_xwT_wmma<<<grid, 32, 0, stream>>>(hv, BI, mlp_w1, MLPN, 0,
                                               mlp_b1, nullptr, Aa, MLPN, 0, BI);
        gemm_xwT_wmma<<<grid, 32, 0, stream>>>(hv, BI, mlp_w1, MLPN, BI,
                                               nullptr, nullptr, Bb, MLPN, 0, BI);
    }

    // 7) borders + fused pairwise scoring into the 321x321 output
    border_kernel<<<(OUTN + 255) / 256, 256, 0, stream>>>(outf);
    {
        dim3 grid(SEQ / 16, SEQ / 16);
        score_kernel<<<grid, 256, 0, stream>>>(Aa, Bb, mlp_w2, mlp_b2, outf);
    }
}